// YoloLayer_40106404610673
// MI455X (gfx1250) — compile-verified
//
#include <hip/hip_runtime.h>
#include <stdint.h>

// YOLO region-box decode for MI455X (gfx1250), TDM edition.
// Bandwidth-bound (~206 MB @ 23.3 TB/s ~= 9 us floor). No matmul structure ->
// WMMA inapplicable. Data movement uses the CDNA5 Tensor Data Mover: one
// tensor_load_to_lds per block stages an 85-channel x 128-cell f32 tile
// (TENSORcnt-tracked), then compute reads LDS with constant offsets only.

typedef unsigned int uv4 __attribute__((ext_vector_type(4)));
typedef int          iv8 __attribute__((ext_vector_type(8)));
typedef int          iv4 __attribute__((ext_vector_type(4)));

namespace {
constexpr int Bn   = 32;
constexpr int An   = 3;
constexpr int Hh   = 76;
constexpr int Ww   = 76;
constexpr int HW   = Hh * Ww;                    // 5776
constexpr int NC   = 80;
constexpr int KCH  = 5 + NC;                     // 85 channels per anchor
constexpr int TILE = 128;                        // hw cells per block (4 wave32)
constexpr int NTILES = (HW + TILE - 1) / TILE;   // 46
constexpr int TAILW  = HW - (NTILES - 1) * TILE; // 16
constexpr size_t BOXES_ELEMS = (size_t)Bn * An * HW * 7;
}

__global__ __launch_bounds__(TILE)
void yolo_decode_tdm(const float* __restrict__ x,
                     const float* __restrict__ thr_p,
                     float* __restrict__ out)
{
    __shared__ float lds[KCH * TILE];            // 85*128*4 = 43520 B

    const int t   = threadIdx.x;
    const int til = blockIdx.x;
    const int a   = blockIdx.y;
    const int b   = blockIdx.z;

    const bool tail  = (til == NTILES - 1);
    const int  tileW = tail ? TAILW : TILE;      // valid cells in this tile

    // ---- wave 0 programs the Tensor Data Mover -------------------------
    // Scalar (per-wave) branch: TDM ignores EXEC, so waves 1..3 must branch
    // around the issue entirely.
    if (__builtin_amdgcn_readfirstlane((int)threadIdx.x) < 32) {
        const float* gptr =
            x + ((size_t)b * (An * KCH) + (size_t)a * KCH) * (size_t)HW
              + (size_t)til * TILE;
        const unsigned long long ga = (unsigned long long)(uintptr_t)gptr;
        const unsigned ldsb = (unsigned)(unsigned long long)(uintptr_t)&lds[0];

        // D# group 0: count=1 | lds_addr | global_addr[56:0] | type=2
        uv4 g0;
        g0.x = 1u;
        g0.y = ldsb;
        g0.z = (unsigned)(ga & 0xffffffffu);
        g0.w = (unsigned)((ga >> 32) & 0x01ffffffu) | 0x80000000u;

        // D# group 1
        unsigned w0 = (2u << 16);                // data_size = 4 bytes
        if (tail) {
            // pad LDS dest: after every 16 DWORDs insert 112 DWORDs so the
            // tail tile keeps the same 128-float row stride as full tiles.
            w0 |= (1u << 20)        // pad_enable
                | (3u << 22)        // pad_interval: 16 DWORDs
                | (111u << 25);     // pad_amount: 112 DWORDs (encode n-1)
        }
        iv8 g1;
        g1[0] = (int)w0;
        g1[1] = (int)(((unsigned)HW & 0xffffu) << 16);        // tensor_dim0 lo16
        g1[2] = (int)((((unsigned)HW >> 16) & 0xffffu)        // tensor_dim0 hi16
                      | ((unsigned)KCH << 16));               // tensor_dim1 lo16
        g1[3] = (int)((unsigned)tileW << 16);                 // dim1 hi16=0 | tile_dim0
        g1[4] = (int)(unsigned)KCH;                           // tile_dim1 | tile_dim2=0
        g1[5] = (int)(unsigned)HW;                            // tensor_dim0_stride lo32
        g1[6] = 0;                                            // stride hi | dim1_stride lo
        g1[7] = 0;                                            // dim1_stride hi

        const iv4 z4 = {0, 0, 0, 0};                          // groups 2/3 unused (2D)
        const iv8 z8 = {0, 0, 0, 0, 0, 0, 0, 0};
        __builtin_amdgcn_tensor_load_to_lds(g0, g1, z4, z4, z8, 0);
        __builtin_amdgcn_s_wait_tensorcnt(0);
    }
    __syncthreads();

    if (t >= tileW) return;                      // no barriers past this point
    const int hw = til * TILE + t;

    // ---- decode (all LDS offsets are compile-time constants) -----------
    const float tx   = lds[0 * TILE + t];
    const float ty   = lds[1 * TILE + t];
    const float tw   = lds[2 * TILE + t];
    const float th2  = lds[3 * TILE + t];
    const float tobj = lds[4 * TILE + t];

    // class logits -> registers (single LDS sweep)
    float v[NC];
    #pragma unroll
    for (int k = 0; k < NC; ++k)
        v[k] = lds[(5 + k) * TILE + t];

    // max + first-occurrence argmax
    float m = v[0];
    int   id = 0;
    #pragma unroll
    for (int k = 1; k < NC; ++k) {
        if (v[k] > m) { m = v[k]; id = k; }
    }
    // softmax max prob = exp(0) / sum = 1 / sum
    float ssum = 0.0f;
    #pragma unroll
    for (int k = 0; k < NC; ++k)
        ssum += __expf(v[k] - m);
    const float cls_max_conf = 1.0f / ssum;

    const float inv_w = 1.0f / (float)Ww;
    const float aw = (a == 0) ? (116.0f / 32.0f)
                   : (a == 1) ? (156.0f / 32.0f) : (373.0f / 32.0f);
    const float ah = (a == 0) ? (90.0f / 32.0f)
                   : (a == 1) ? (198.0f / 32.0f) : (326.0f / 32.0f);

    const int gx = hw % Ww;
    const int gy = hw / Ww;

    const float xs  = (1.0f / (1.0f + __expf(-tx)) + (float)gx) * inv_w;
    const float ys  = (1.0f / (1.0f + __expf(-ty)) + (float)gy) * inv_w;
    const float ws  = __expf(tw)  * aw * inv_w;
    const float hs  = __expf(th2) * ah * inv_w;
    const float det = 1.0f / (1.0f + __expf(-tobj));

    // ---- outputs: boxes [B,A,HW,7] then keep [B,A,HW] ------------------
    const size_t cell = ((size_t)(b * An + a) * HW + (size_t)hw);
    float* bp = out + cell * 7;
    bp[0] = xs;
    bp[1] = ys;
    bp[2] = ws;
    bp[3] = hs;
    bp[4] = det;
    bp[5] = cls_max_conf;
    bp[6] = (float)id;

    const float thr = thr_p[0];
    out[BOXES_ELEMS + cell] = (det > thr) ? 1.0f : 0.0f;
}

extern "C" void kernel_launch(void* const* d_in, const int* in_sizes, int n_in,
                              void* d_out, int out_size, void* d_ws, size_t ws_size,
                              hipStream_t stream)
{
    (void)in_sizes; (void)n_in; (void)out_size; (void)d_ws; (void)ws_size;
    const float* x     = (const float*)d_in[0];
    const float* thr_p = (const float*)d_in[1];
    float*       out   = (float*)d_out;

    dim3 grid(NTILES, An, Bn);   // 46 x 3 x 32
    dim3 block(TILE);            // 128 threads = 4 wave32
    hipLaunchKernelGGL(yolo_decode_tdm, grid, block, 0, stream,
                       x, thr_p, out);
}